// split_stride_conv_75960791597929
// MI455X (gfx1250) — compile-verified
//
#include <hip/hip_runtime.h>

// Implicit-GEMM formulation of the circular-pad 3x3 conv (1 -> 64 channels)
// for MI455X (gfx1250, wave32):  D(64 x Npix) = W(64x9) * X(9 x Npix) + bias,
// K padded 9->12, executed as 3 x V_WMMA_F32_16X16X4_F32 per 16x16 tile.
// Store-bandwidth bound: 368.6 MB output -> ~16us roofline at 23.3 TB/s.
// Round 2: branchless per-half tap selection (v_cndmask instead of
// exec-masked branches) so all loads issue in uniform clauses.

typedef __attribute__((ext_vector_type(2))) float v2f;
typedef __attribute__((ext_vector_type(8))) float v8f;

#define BATCH 4
#define CHAN  64
#define HH    600
#define WW    600
#define HWSZ  (HH * WW)          // 360000
#define NPIX  (BATCH * HWSZ)     // 1440000
#define WAVES_PER_BLOCK 8
#define NTILES (NPIX / 16)       // 90000 (exact)

__global__ __launch_bounds__(256)
void conv3x3_wrap_wmma(const float* __restrict__ x,
                       const float* __restrict__ wgt,   // [64][9]
                       const float* __restrict__ bias,  // [64]
                       float* __restrict__ out)         // [4][64][600][600]
{
    const int lane = threadIdx.x & 31;
    const int wave = threadIdx.x >> 5;
    const int tile = blockIdx.x * WAVES_PER_BLOCK + wave;  // grid sized exactly
    const bool hi  = (lane & 16) != 0;  // lanes 16-31 supply K=2,3 of each step
    const int n    = lane & 15;         // pixel column (B/D) == channel row (A)

    // ---- decompose this lane's pixel (B-matrix column / D-matrix column) ----
    const int p   = tile * 16 + n;
    const int b   = p / HWSZ;
    const int rem = p - b * HWSZ;
    const int y   = rem / WW;
    const int xx  = rem - y * WW;

    // circular (wrap) neighbors
    const int ym1 = (y  == 0     ) ? HH - 1 : y  - 1;
    const int yp1 = (y  == HH - 1) ? 0      : y  + 1;
    const int xm1 = (xx == 0     ) ? WW - 1 : xx - 1;
    const int xp1 = (xx == WW - 1) ? 0      : xx + 1;

    const float* xb = x + b * HWSZ;

    // ---- B fragments: 4x16 f32 per K-step, ISA layout:
    // VGPR0 = {K=0 | K=2}, VGPR1 = {K=1 | K=3} across lane halves.
    // Taps t = r*3+s over rows {ym1,y,yp1} x cols {xm1,xx,xp1}; taps 9..11 = 0.
    // Branchless: select tap coordinates per lane half, loads unconditional.
    const int r0 = ym1;                // lo: tap0 (ym1,xm1)   hi: tap2 (ym1,xp1)
    const int c0 = hi ? xp1 : xm1;
    const int r1 = hi ? y   : ym1;     // lo: tap1 (ym1,xx)    hi: tap3 (y,xm1)
    const int c1 = hi ? xm1 : xx;
    const int r2 = hi ? yp1 : y;       // lo: tap4 (y,xx)      hi: tap6 (yp1,xm1)
    const int c2 = hi ? xm1 : xx;
    const int r3 = hi ? yp1 : y;       // lo: tap5 (y,xp1)     hi: tap7 (yp1,xx)
    const int c3 = hi ? xx  : xp1;

    v2f bf0, bf1, bf2;
    bf0.x = xb[r0 * WW + c0];
    bf0.y = xb[r1 * WW + c1];
    bf1.x = xb[r2 * WW + c2];
    bf1.y = xb[r3 * WW + c3];
    const float tap8 = xb[yp1 * WW + xp1];  // loaded by all lanes (valid addr)
    bf2.x = hi ? 0.0f : tap8;               // v_cndmask, no exec branch
    bf2.y = 0.0f;                           // taps 9..11 are K-padding

    // A-fragment tap selectors per lane half (A layout mirrors B)
    const int t0 = hi ? 2 : 0;
    const int t1 = hi ? 3 : 1;
    const int t2 = hi ? 6 : 4;
    const int t3 = hi ? 7 : 5;

    float* outLane = out + (size_t)b * (CHAN * HWSZ) + (size_t)(y * WW + xx);

    // ---- all 4 channel groups reuse the same B fragments ----
#pragma unroll
    for (int g = 0; g < 4; ++g) {
        const float* wrow = wgt + (g * 16 + n) * 9;   // A row = channel g*16+n
        v2f a0, a1, a2;
        a0.x = wrow[t0]; a0.y = wrow[t1];
        a1.x = wrow[t2]; a1.y = wrow[t3];
        const float w8 = wrow[8];                     // unconditional load
        a2.x = hi ? 0.0f : w8;                        // select, no branch
        a2.y = 0.0f;

        // C/D layout: lane n, VGPR i -> channel cbase+i, pixel n&15.
        // Fold bias into accumulator init.
        const int cbase = g * 16 + (hi ? 8 : 0);
        v8f acc;
#pragma unroll
        for (int i = 0; i < 8; ++i) acc[i] = bias[cbase + i];

        // 8-arg form: (neg_a, A, neg_b, B, c_mod, C, reuse_a, reuse_b)
        acc = __builtin_amdgcn_wmma_f32_16x16x4_f32(false, a0, false, bf0,
                                                    (short)0, acc, false, false);
        acc = __builtin_amdgcn_wmma_f32_16x16x4_f32(false, a1, false, bf1,
                                                    (short)0, acc, false, false);
        acc = __builtin_amdgcn_wmma_f32_16x16x4_f32(false, a2, false, bf2,
                                                    (short)0, acc, false, false);

        // Streaming stores: output (369 MB) exceeds L2 (192 MB) and is write-once.
        float* op = outLane + (size_t)cbase * HWSZ;
#pragma unroll
        for (int i = 0; i < 8; ++i)
            __builtin_nontemporal_store(acc[i], op + (size_t)i * HWSZ);
    }
}

extern "C" void kernel_launch(void* const* d_in, const int* in_sizes, int n_in,
                              void* d_out, int out_size, void* d_ws, size_t ws_size,
                              hipStream_t stream)
{
    const float* x    = (const float*)d_in[0];   // [4][1][600][600]
    const float* wgt  = (const float*)d_in[1];   // [64][1][3][3]
    const float* bias = (const float*)d_in[2];   // [64]
    float* out = (float*)d_out;                  // [4][64][600][600]

    dim3 block(256);                              // 8 waves (wave32)
    dim3 grid(NTILES / WAVES_PER_BLOCK);          // 90000/8 = 11250, exact
    conv3x3_wrap_wmma<<<grid, block, 0, stream>>>(x, wgt, bias, out);

    (void)in_sizes; (void)n_in; (void)out_size; (void)d_ws; (void)ws_size;
}